// SentencePredictionHead_48060684043038
// MI455X (gfx1250) — compile-verified
//
#include <hip/hip_runtime.h>
#include <math.h>

// ---------------------------------------------------------------------------
// Problem constants (static in the reference)
// ---------------------------------------------------------------------------
#define NROWS   32768       // B*L
#define HDIM    1024
#define KSEL    8192
#define LN_EPS  1e-12f
#define PD_EPS  1e-6f

typedef __attribute__((ext_vector_type(2))) float v2f;
typedef __attribute__((ext_vector_type(8))) float v8f;

// ---------------------------------------------------------------------------
// CDNA5 async global->LDS support (guarded; safe fallback if builtin absent)
// Builtin signature (from clang diagnostics):
//   void __builtin_amdgcn_global_load_async_to_lds_b128(
//       v4i AS1* src, v4i AS3* dst, imm int offset, imm int cpol)
// ---------------------------------------------------------------------------
#if defined(__has_builtin)
# if __has_builtin(__builtin_amdgcn_global_load_async_to_lds_b128)
#  define USE_ASYNC_LDS 1
# endif
#endif

typedef int v4i_t __attribute__((vector_size(4 * sizeof(int))));
typedef __attribute__((address_space(1))) v4i_t as1_v4i_t;
typedef __attribute__((address_space(3))) v4i_t as3_v4i_t;

__device__ __forceinline__ as1_v4i_t* as_global_v4i(const void* p) {
    // global flat addresses are identity-mapped to AS1
    return (as1_v4i_t*)(unsigned long long)(uintptr_t)p;
}
__device__ __forceinline__ as3_v4i_t* as_lds_v4i(const void* p) {
    // ISA 10.2: LDS address is the low 32 bits of the flat address
    return (as3_v4i_t*)(unsigned int)(uintptr_t)p;
}
__device__ __forceinline__ void wait_async0() {
#if defined(__has_builtin) && __has_builtin(__builtin_amdgcn_s_wait_asynccnt)
    __builtin_amdgcn_s_wait_asynccnt(0);
#else
    asm volatile("s_wait_asynccnt 0x0" ::: "memory");
#endif
}

// ---------------------------------------------------------------------------
// 1) Order-preserving stream compaction of the boolean mask -> idx[KSEL]
// ---------------------------------------------------------------------------
__global__ __launch_bounds__(1024)
void compact_mask_kernel(const unsigned char* __restrict__ mask,
                         int* __restrict__ idx) {
    __shared__ int cnt[1024];
    __shared__ int off[1024];
    const int t = threadIdx.x;
    const int per = NROWS / 1024;           // 32 elements per thread
    const int base = t * per;
    int c = 0;
    for (int e = 0; e < per; ++e) c += (mask[base + e] != 0);
    cnt[t] = c;
    __syncthreads();
    if (t == 0) {                            // tiny serial exclusive scan
        int run = 0;
        for (int i = 0; i < 1024; ++i) { off[i] = run; run += cnt[i]; }
    }
    __syncthreads();
    int pos = off[t];
    for (int e = 0; e < per; ++e)
        if (mask[base + e] != 0) idx[pos++] = base + e;
}

// ---------------------------------------------------------------------------
// 2+4) Generic C = A * B^T (+bias) GEMM on fp32 WMMA (v_wmma_f32_16x16x4_f32)
//      Double-buffered LDS, async global->LDS pipeline when available.
//      Block tile 128(M) x 64(N), K-tile 32. 256 threads = 8 waves,
//      each wave owns a 32x32 region = 2x2 WMMA accumulators.
// ---------------------------------------------------------------------------
#define TM 128
#define TN 64
#define TK 32
#define LDST (TK + 4)   // stride 36 floats: 16B-aligned rows, conflict-free frags

__global__ __launch_bounds__(256)
void gemm_nt_wmma_kernel(const float* __restrict__ A, const int* __restrict__ aidx, int lda,
                         const float* __restrict__ Bm, const int* __restrict__ bidx, int ldb,
                         float* __restrict__ C, long ldc,
                         const float* __restrict__ bias, int Kdim) {
    __shared__ __align__(16) float As[2][TM][LDST];
    __shared__ __align__(16) float Bs[2][TN][LDST];

    const int t    = threadIdx.x;
    const int lane = t & 31;
    const int wave = t >> 5;
    const int half = lane >> 4;      // 0: lanes 0-15, 1: lanes 16-31
    const int l16  = lane & 15;
    const int wm   = (wave & 3) * 32;   // wave M offset inside block tile
    const int wn   = (wave >> 2) * 32;  // wave N offset inside block tile
    const int mBase = blockIdx.y * TM;
    const int nBase = blockIdx.x * TN;

    // ---- resolve gathered row base pointers ONCE (hoisted out of K loop) ----
    const float* aSrc[4]; int aRow[4], aCol[4];
    #pragma unroll
    for (int i = 0; i < 4; ++i) {
        const int lin = t + i * 256;
        aRow[i] = lin >> 3;                  // 8 x float4 per 32-float row
        aCol[i] = (lin & 7) << 2;
        const int gr = aidx ? aidx[mBase + aRow[i]] : (mBase + aRow[i]);
        aSrc[i] = A + (size_t)gr * lda + aCol[i];
    }
    const float* bSrc[2]; int bRow[2], bCol[2];
    #pragma unroll
    for (int i = 0; i < 2; ++i) {
        const int lin = t + i * 256;
        bRow[i] = lin >> 3;
        bCol[i] = (lin & 7) << 2;
        const int gr = bidx ? bidx[nBase + bRow[i]] : (nBase + bRow[i]);
        bSrc[i] = Bm + (size_t)gr * ldb + bCol[i];
    }

    v8f zero = {};
    v8f acc[2][2];
    acc[0][0] = zero; acc[0][1] = zero; acc[1][0] = zero; acc[1][1] = zero;

    const int ntile = Kdim / TK;

#ifdef USE_ASYNC_LDS
    // ---- prologue: async-issue K-tile 0 into buffer 0 ----
    #pragma unroll
    for (int i = 0; i < 4; ++i)
        __builtin_amdgcn_global_load_async_to_lds_b128(
            as_global_v4i(aSrc[i]), as_lds_v4i(&As[0][aRow[i]][aCol[i]]), 0, 0);
    #pragma unroll
    for (int i = 0; i < 2; ++i)
        __builtin_amdgcn_global_load_async_to_lds_b128(
            as_global_v4i(bSrc[i]), as_lds_v4i(&Bs[0][bRow[i]][bCol[i]]), 0, 0);
#endif

    for (int tt = 0; tt < ntile; ++tt) {
        const int buf = tt & 1;

#ifdef USE_ASYNC_LDS
        // wait for my async fills of As[buf]/Bs[buf]; barrier makes them
        // globally visible and guarantees everyone is done reading buf^1.
        wait_async0();
        __syncthreads();
        if (tt + 1 < ntile) {
            const int nb = buf ^ 1;
            const int k1 = (tt + 1) * TK;
            #pragma unroll
            for (int i = 0; i < 4; ++i)
                __builtin_amdgcn_global_load_async_to_lds_b128(
                    as_global_v4i(aSrc[i] + k1),
                    as_lds_v4i(&As[nb][aRow[i]][aCol[i]]), 0, 0);
            #pragma unroll
            for (int i = 0; i < 2; ++i)
                __builtin_amdgcn_global_load_async_to_lds_b128(
                    as_global_v4i(bSrc[i] + k1),
                    as_lds_v4i(&Bs[nb][bRow[i]][bCol[i]]), 0, 0);
        }
#else
        // synchronous fallback: register-staged copy into buffer
        const int k0 = tt * TK;
        float4 va[4], vb[2];
        #pragma unroll
        for (int i = 0; i < 4; ++i)
            va[i] = *reinterpret_cast<const float4*>(aSrc[i] + k0);
        #pragma unroll
        for (int i = 0; i < 2; ++i)
            vb[i] = *reinterpret_cast<const float4*>(bSrc[i] + k0);
        __syncthreads();   // everyone done reading this buffer (2 tiles ago)
        #pragma unroll
        for (int i = 0; i < 4; ++i) {
            As[buf][aRow[i]][aCol[i] + 0] = va[i].x;
            As[buf][aRow[i]][aCol[i] + 1] = va[i].y;
            As[buf][aRow[i]][aCol[i] + 2] = va[i].z;
            As[buf][aRow[i]][aCol[i] + 3] = va[i].w;
        }
        #pragma unroll
        for (int i = 0; i < 2; ++i) {
            Bs[buf][bRow[i]][bCol[i] + 0] = vb[i].x;
            Bs[buf][bRow[i]][bCol[i] + 1] = vb[i].y;
            Bs[buf][bRow[i]][bCol[i] + 2] = vb[i].z;
            Bs[buf][bRow[i]][bCol[i] + 3] = vb[i].w;
        }
        __syncthreads();
#endif

        // ---- 8 k-steps of 4, 4 fp32 WMMAs each ----
        #pragma unroll
        for (int kk = 0; kk < TK; kk += 4) {
            const int ka = kk + (half << 1);     // K = 2*half + {0,1}
            v2f a0, a1, b0, b1;
            a0.x = As[buf][wm      + l16][ka]; a0.y = As[buf][wm      + l16][ka + 1];
            a1.x = As[buf][wm + 16 + l16][ka]; a1.y = As[buf][wm + 16 + l16][ka + 1];
            b0.x = Bs[buf][wn      + l16][ka]; b0.y = Bs[buf][wn      + l16][ka + 1];
            b1.x = Bs[buf][wn + 16 + l16][ka]; b1.y = Bs[buf][wn + 16 + l16][ka + 1];
            acc[0][0] = __builtin_amdgcn_wmma_f32_16x16x4_f32(
                false, a0, false, b0, (short)0, acc[0][0], false, false);
            acc[0][1] = __builtin_amdgcn_wmma_f32_16x16x4_f32(
                false, a0, false, b1, (short)0, acc[0][1], false, false);
            acc[1][0] = __builtin_amdgcn_wmma_f32_16x16x4_f32(
                false, a1, false, b0, (short)0, acc[1][0], false, false);
            acc[1][1] = __builtin_amdgcn_wmma_f32_16x16x4_f32(
                false, a1, false, b1, (short)0, acc[1][1], false, false);
        }
    }

    // ---- epilogue: C/D layout -> M = r + 8*half, N = l16 ----
    #pragma unroll
    for (int mt = 0; mt < 2; ++mt) {
        #pragma unroll
        for (int nt = 0; nt < 2; ++nt) {
            const int n = nBase + wn + nt * 16 + l16;
            const float bv = bias ? bias[n] : 0.0f;
            #pragma unroll
            for (int r = 0; r < 8; ++r) {
                const int m = mBase + wm + mt * 16 + r + half * 8;
                C[(size_t)m * ldc + n] = acc[mt][nt][r] + bv;
            }
        }
    }
}

// ---------------------------------------------------------------------------
// 3) In-place LayerNorm over H=1024 (one block per row)
// ---------------------------------------------------------------------------
__global__ __launch_bounds__(256)
void layernorm_kernel(float* __restrict__ h,
                      const float* __restrict__ gamma,
                      const float* __restrict__ beta) {
    __shared__ float s1[256], s2[256];
    const int row = blockIdx.x;
    const int t = threadIdx.x;
    float* p = h + (size_t)row * HDIM;
    const float x0 = p[t], x1 = p[t + 256], x2 = p[t + 512], x3 = p[t + 768];
    s1[t] = x0 + x1 + x2 + x3;
    s2[t] = x0 * x0 + x1 * x1 + x2 * x2 + x3 * x3;
    __syncthreads();
    for (int o = 128; o > 0; o >>= 1) {
        if (t < o) { s1[t] += s1[t + o]; s2[t] += s2[t + o]; }
        __syncthreads();
    }
    const float mu  = s1[0] * (1.0f / HDIM);
    const float var = s2[0] * (1.0f / HDIM) - mu * mu;
    const float rs  = rsqrtf(var + LN_EPS);
    p[t]       = (x0 - mu) * rs * gamma[t]       + beta[t];
    p[t + 256] = (x1 - mu) * rs * gamma[t + 256] + beta[t + 256];
    p[t + 512] = (x2 - mu) * rs * gamma[t + 512] + beta[t + 512];
    p[t + 768] = (x3 - mu) * rs * gamma[t + 768] + beta[t + 768];
}

// ---------------------------------------------------------------------------
// 5) Per-row softmax losses. Whole 8192-float row staged once in 32KB LDS.
// ---------------------------------------------------------------------------
__global__ __launch_bounds__(256)
void softmax_loss_kernel(const float* __restrict__ logits,
                         float* __restrict__ pel_dist,
                         float* __restrict__ pel_prod) {
    __shared__ float rowbuf[KSEL];
    __shared__ float red[256];
    const int i = blockIdx.x;
    const int t = threadIdx.x;
    const float* r = logits + (size_t)i * KSEL;

    float mx = -INFINITY;
    for (int j = t; j < KSEL; j += 256) {
        const float v = r[j];
        rowbuf[j] = v;
        mx = fmaxf(mx, v);
    }
    red[t] = mx; __syncthreads();
    for (int o = 128; o > 0; o >>= 1) {
        if (t < o) red[t] = fmaxf(red[t], red[t + o]);
        __syncthreads();
    }
    mx = red[0];
    __syncthreads();

    float s = 0.0f;
    for (int j = t; j < KSEL; j += 256) s += expf(rowbuf[j] - mx);
    red[t] = s; __syncthreads();
    for (int o = 128; o > 0; o >>= 1) {
        if (t < o) red[t] += red[t + o];
        __syncthreads();
    }
    s = red[0];
    __syncthreads();

    const float inv = 1.0f / s;
    float d = 0.0f;
    for (int j = t; j < KSEL; j += 256) {
        const float pr = expf(rowbuf[j] - mx) * inv;
        const float e  = (j == i) ? 1.0f : 0.0f;
        d += fabsf(pr - e + PD_EPS);
    }
    red[t] = d; __syncthreads();
    for (int o = 128; o > 0; o >>= 1) {
        if (t < o) red[t] += red[t + o];
        __syncthreads();
    }
    if (t == 0) {
        pel_dist[i] = red[0];
        const float lp = (rowbuf[i] - mx) - logf(s);   // log_softmax diagonal
        pel_prod[i] = -lp;
    }
}

// ---------------------------------------------------------------------------
// 6) Final scalar reductions: sums/K and ddof=1 variances
// ---------------------------------------------------------------------------
__global__ __launch_bounds__(1024)
void final_reduce_kernel(const float* __restrict__ pel_dist,
                         const float* __restrict__ pel_prod,
                         float* __restrict__ out4) {
    __shared__ float sd[1024], sp[1024], qd[1024], qp[1024];
    const int t = threadIdx.x;
    float a = 0, b = 0, c = 0, d = 0;
    for (int j = t; j < KSEL; j += 1024) {
        const float x = pel_dist[j], y = pel_prod[j];
        a += x; c += x * x; b += y; d += y * y;
    }
    sd[t] = a; sp[t] = b; qd[t] = c; qp[t] = d;
    __syncthreads();
    for (int o = 512; o > 0; o >>= 1) {
        if (t < o) { sd[t] += sd[t + o]; sp[t] += sp[t + o];
                     qd[t] += qd[t + o]; qp[t] += qp[t + o]; }
        __syncthreads();
    }
    if (t == 0) {
        const float denom = (float)KSEL + 1e-5f;
        const float sumd = sd[0], sump = sp[0];
        out4[0] = sumd / denom;                                      // loss_distance
        out4[1] = sump / denom;                                      // loss_product
        out4[2] = (qd[0] - sumd * sumd / (float)KSEL) / (float)(KSEL - 1);
        out4[3] = (qp[0] - sump * sump / (float)KSEL) / (float)(KSEL - 1);
    }
}

// ---------------------------------------------------------------------------
// Host-side launcher
// ---------------------------------------------------------------------------
extern "C" void kernel_launch(void* const* d_in, const int* in_sizes, int n_in,
                              void* d_out, int out_size, void* d_ws, size_t ws_size,
                              hipStream_t stream) {
    (void)in_sizes; (void)n_in; (void)out_size; (void)ws_size;
    const float*         msp   = (const float*)d_in[0];          // [32768,1024]
    const float*         lab   = (const float*)d_in[1];          // [32768,1024]
    const unsigned char* mask  = (const unsigned char*)d_in[2];  // [32768] bool
    const float*         dw    = (const float*)d_in[3];          // [1024,1024]
    const float*         db    = (const float*)d_in[4];          // [1024]
    const float*         gamma = (const float*)d_in[5];          // [1024]
    const float*         beta  = (const float*)d_in[6];          // [1024]

    float* out = (float*)d_out;
    float* logits   = out;                               // [K,K]
    float* pel_dist = out + (size_t)KSEL * KSEL;         // [K]
    float* pel_prod = pel_dist + KSEL;                   // [K]
    float* scalars  = pel_prod + KSEL;                   // [4]

    // workspace: idx (32KB, 256B-aligned region) then h [K,H] fp32 (32MB)
    int*   idx = (int*)d_ws;
    float* h   = (float*)((char*)d_ws + (((size_t)KSEL * sizeof(int) + 255) & ~(size_t)255));

    // 1) compaction
    compact_mask_kernel<<<1, 1024, 0, stream>>>(mask, idx);

    // 2) h = gather(preds) @ W^T + b   [8192 x 1024]
    gemm_nt_wmma_kernel<<<dim3(HDIM / TN, KSEL / TM), 256, 0, stream>>>(
        msp, idx, HDIM,          // A gathered
        dw, nullptr, HDIM,       // B dense
        h, (long)HDIM, db, HDIM);

    // 3) LayerNorm in place
    layernorm_kernel<<<KSEL, 256, 0, stream>>>(h, gamma, beta);

    // 4) logits = h_ln @ gather(labels)^T   [8192 x 8192] -> d_out
    gemm_nt_wmma_kernel<<<dim3(KSEL / TN, KSEL / TM), 256, 0, stream>>>(
        h, nullptr, HDIM,        // A dense
        lab, idx, HDIM,          // B gathered
        logits, (long)KSEL, nullptr, HDIM);

    // 5) per-row softmax losses
    softmax_loss_kernel<<<KSEL, 256, 0, stream>>>(logits, pel_dist, pel_prod);

    // 6) scalar reductions
    final_reduce_kernel<<<1, 1024, 0, stream>>>(pel_dist, pel_prod, scalars);
}